// HGTLayerwithEdgeFeat_71279277244883
// MI455X (gfx1250) — compile-verified
//
#include <hip/hip_runtime.h>
#include <hip/hip_bf16.h>

typedef __attribute__((ext_vector_type(2))) float v2f;
typedef __attribute__((ext_vector_type(8))) float v8f;

#define DMODEL 256

// ---------------------------------------------------------------------------
// CDNA5 async global->LDS copy (16B), tracked by ASYNCcnt.
// Low 32 bits of a generic pointer to LDS == wave-relative LDS byte offset.
// ---------------------------------------------------------------------------
__device__ __forceinline__ void async_copy16(void* lds_dst, const void* gsrc) {
  unsigned l = (unsigned)(unsigned long long)lds_dst;
  asm volatile("global_load_async_to_lds_b128 %0, %1, off"
               :: "v"(l), "v"(gsrc) : "memory");
}
__device__ __forceinline__ void wait_async_le10() {
  asm volatile("s_wait_asynccnt 0xa" ::: "memory");
}
__device__ __forceinline__ void wait_async_le0() {
  asm volatile("s_wait_asynccnt 0x0" ::: "memory");
}

// ---------------------------------------------------------------------------
// Weight prep: fold per-head relation transform into projection weight,
// stored OUTPUT-major so the GEMM's B fragments are K-adjacent in LDS:
//   Wt[h*64+e, c] = sum_d W[h*64+d, c] * T[h, d, e]
//   bt[h*64+e]    = sum_d b[h*64+d]    * T[h, d, e]
// grid.x = 256 (o = h*64+e), block = 256 (c)  -> coalesced reads & writes
// ---------------------------------------------------------------------------
__global__ __launch_bounds__(256) void prep_combined(
    const float* __restrict__ W, const float* __restrict__ b,
    const float* __restrict__ T, float* __restrict__ Wt, float* __restrict__ bt) {
  int o = blockIdx.x;
  int c = threadIdx.x;
  int h = o >> 6, e = o & 63;
  const float* Th = T + (size_t)h * 64 * 64;
  float s = 0.f;
  for (int d = 0; d < 64; ++d)
    s += W[(size_t)(h * 64 + d) * DMODEL + c] * Th[d * 64 + e];
  Wt[(size_t)o * DMODEL + c] = s;
  if (c == 0) {
    float sb = 0.f;
    for (int d = 0; d < 64; ++d)
      sb += b[h * 64 + d] * Th[d * 64 + e];
    bt[o] = sb;
  }
}

__global__ void fill_f32(float* __restrict__ p, float v, int n) {
  int i = blockIdx.x * blockDim.x + threadIdx.x;
  if (i < n) p[i] = v;
}

// ---------------------------------------------------------------------------
// GEMM: C[M x 256] = A[M x 256] @ Wt^T + bias   (Wt is [256 out][256 in])
// fp32 WMMA 16x16x4. Block = 256 threads (8 waves). Block tile 64(M) x 256(N).
// Waves 2(M) x 4(N); each wave: 32x64 tile = 2x4 v8f accumulators.
// Double-buffered LDS filled by async global->LDS B128 copies.
// grid.x = M/64
// ---------------------------------------------------------------------------
__global__ __launch_bounds__(256) void gemm_wmma_f32(
    const float* __restrict__ A, const float* __restrict__ Wt,
    const float* __restrict__ bias, float* __restrict__ C, int M) {
  __shared__ float As[2][64][36];    // [buf][m][k]  (+4 pad keeps 16B align)
  __shared__ float Bs[2][256][36];   // [buf][n][k]

  int tid = threadIdx.x;
  int lane = tid & 31;
  int wave = tid >> 5;
  int m0 = (wave & 1) * 32;          // wave row: 0 or 32
  int n0 = (wave >> 1) * 64;         // wave col: 0,64,128,192
  int blockM = blockIdx.x * 64;

  // loader mapping: A rows of 32 floats, 8 per thread; B one row per thread
  int ar = tid >> 2;                 // 0..63
  int ac = (tid & 3) * 8;            // 0,8,16,24

  v8f zf = {0.f, 0.f, 0.f, 0.f, 0.f, 0.f, 0.f, 0.f};
  v8f acc[2][4] = {{zf, zf, zf, zf}, {zf, zf, zf, zf}};

  int mr = lane & 15;
  int khalf = (lane >> 4) * 2;       // lanes 0-15 -> K{0,1}; 16-31 -> K{2,3}
  int ncol = lane & 15;

  auto issue = [&](int buf, int k0) {
    const float* ag = A + (size_t)(blockM + ar) * DMODEL + k0 + ac;
    async_copy16(&As[buf][ar][ac], ag);
    async_copy16(&As[buf][ar][ac + 4], ag + 4);
    const float* bg = Wt + (size_t)tid * DMODEL + k0;
#pragma unroll
    for (int i = 0; i < 8; ++i)
      async_copy16(&Bs[buf][tid][i * 4], bg + i * 4);
  };

  issue(0, 0);
  for (int c = 0; c < 8; ++c) {
    int buf = c & 1;
    if (c < 7) {
      issue(buf ^ 1, (c + 1) * 32);  // prefetch next chunk into other buffer
      wait_async_le10();             // current buffer's 10 copies are done
    } else {
      wait_async_le0();
    }
    __syncthreads();

#pragma unroll
    for (int kk = 0; kk < 32; kk += 4) {
      v2f a[2], b[4];
#pragma unroll
      for (int i = 0; i < 2; ++i)
        a[i] = *(const v2f*)&As[buf][m0 + i * 16 + mr][kk + khalf];
#pragma unroll
      for (int j = 0; j < 4; ++j)
        b[j] = *(const v2f*)&Bs[buf][n0 + j * 16 + ncol][kk + khalf];
#pragma unroll
      for (int i = 0; i < 2; ++i)
#pragma unroll
        for (int j = 0; j < 4; ++j)
          acc[i][j] = __builtin_amdgcn_wmma_f32_16x16x4_f32(
              false, a[i], false, b[j], (short)0, acc[i][j], false, false);
    }
    __syncthreads();
  }

  int mhi = (lane >> 4) * 8;
#pragma unroll
  for (int i = 0; i < 2; ++i) {
#pragma unroll
    for (int j = 0; j < 4; ++j) {
      int col = n0 + j * 16 + ncol;
      float bv = bias[col];
#pragma unroll
      for (int r = 0; r < 8; ++r) {
        int row = blockM + m0 + i * 16 + mhi + r;
        C[(size_t)row * DMODEL + col] = acc[i][j][r] + bv;
      }
    }
  }
}

// ---------------------------------------------------------------------------
// Edge kernels: wave per edge; lane covers 8 contiguous elems, head = lane/8.
// ---------------------------------------------------------------------------
__device__ inline void atomicMaxF(float* addr, float val) {
  unsigned int* ua = (unsigned int*)addr;
  unsigned int old = *ua;
  while (__uint_as_float(old) < val) {
    unsigned int assumed = old;
    old = atomicCAS(ua, assumed, __float_as_uint(val));
    if (old == assumed) break;
  }
}

__global__ __launch_bounds__(256) void edge_score(
    const float* __restrict__ Q, const float* __restrict__ Kt,
    const int* __restrict__ src, const int* __restrict__ dst,
    const float* __restrict__ pri, float* __restrict__ score,
    float* __restrict__ smax, int nE) {
  int e = blockIdx.x * 8 + (threadIdx.x >> 5);
  if (e >= nE) return;
  int lane = threadIdx.x & 31;
  int s = src[e], d = dst[e];
  const float4* qp = (const float4*)(Q + (size_t)d * DMODEL) + lane * 2;
  const float4* kp = (const float4*)(Kt + (size_t)s * DMODEL) + lane * 2;
  float4 q0 = qp[0], q1 = qp[1];
  float4 k0 = kp[0], k1 = kp[1];
  float p = q0.x * k0.x + q0.y * k0.y + q0.z * k0.z + q0.w * k0.w +
            q1.x * k1.x + q1.y * k1.y + q1.z * k1.z + q1.w * k1.w;
  p += __shfl_xor(p, 1, 32);
  p += __shfl_xor(p, 2, 32);
  p += __shfl_xor(p, 4, 32);
  int h = lane >> 3;
  if ((lane & 7) == 0) {
    float sc = p * pri[h] * 0.125f;  // 1/sqrt(64)
    score[(size_t)e * 4 + h] = sc;
    atomicMaxF(&smax[(size_t)d * 4 + h], sc);
  }
}

__global__ __launch_bounds__(256) void edge_exp(
    const int* __restrict__ dst, float* __restrict__ score,
    const float* __restrict__ smax, float* __restrict__ den, int nE) {
  int i = blockIdx.x * 256 + threadIdx.x;
  if (i >= nE * 4) return;
  int e = i >> 2, h = i & 3;
  int d = dst[e];
  float ex = expf(score[i] - smax[(size_t)d * 4 + h]);
  score[i] = ex;
  atomicAdd(&den[(size_t)d * 4 + h], ex);
}

__global__ __launch_bounds__(256) void edge_agg(
    const float* __restrict__ Vt, const int* __restrict__ src,
    const int* __restrict__ dst, const float* __restrict__ ex,
    const float* __restrict__ den, float* __restrict__ agg, int nE) {
  int e = blockIdx.x * 8 + (threadIdx.x >> 5);
  if (e >= nE) return;
  int lane = threadIdx.x & 31;
  int s = src[e], d = dst[e];
  int h = lane >> 3;
  float attn = ex[(size_t)e * 4 + h] / den[(size_t)d * 4 + h];
  const float4* vp = (const float4*)(Vt + (size_t)s * DMODEL) + lane * 2;
  float4 v0 = vp[0], v1 = vp[1];
  float* ap = agg + (size_t)d * DMODEL + lane * 8;
  atomicAdd(ap + 0, v0.x * attn);
  atomicAdd(ap + 1, v0.y * attn);
  atomicAdd(ap + 2, v0.z * attn);
  atomicAdd(ap + 3, v0.w * attn);
  atomicAdd(ap + 4, v1.x * attn);
  atomicAdd(ap + 5, v1.y * attn);
  atomicAdd(ap + 6, v1.z * attn);
  atomicAdd(ap + 7, v1.w * attn);
}

// ---------------------------------------------------------------------------
// out = LN(agg*scale*alpha + P*(1-alpha)) * g + be ;  alpha = sigmoid(skip)
// one block (256 threads) per row
// ---------------------------------------------------------------------------
__global__ __launch_bounds__(256) void finalize_ln(
    const float* __restrict__ agg, float aggScale,
    const float* __restrict__ P, const float* __restrict__ skip,
    const float* __restrict__ g, const float* __restrict__ be,
    float* __restrict__ out) {
  __shared__ float red[32];
  int n = blockIdx.x;
  int t = threadIdx.x;
  float alpha = 1.f / (1.f + expf(-skip[0]));
  float x = agg[(size_t)n * DMODEL + t] * aggScale * alpha +
            P[(size_t)n * DMODEL + t] * (1.f - alpha);
  float s = x, s2 = x * x;
#pragma unroll
  for (int m = 16; m >= 1; m >>= 1) {
    s += __shfl_xor(s, m, 32);
    s2 += __shfl_xor(s2, m, 32);
  }
  int wave = t >> 5, lane = t & 31;
  if (lane == 0) { red[wave] = s; red[8 + wave] = s2; }
  __syncthreads();
  if (t == 0) {
    float ts = 0.f, ts2 = 0.f;
    for (int w = 0; w < 8; ++w) { ts += red[w]; ts2 += red[8 + w]; }
    red[16] = ts * (1.f / DMODEL);
    red[17] = ts2 * (1.f / DMODEL);
  }
  __syncthreads();
  float mean = red[16];
  float var = red[17] - mean * mean;
  out[(size_t)n * DMODEL + t] = (x - mean) * rsqrtf(var + 1e-5f) * g[t] + be[t];
}

// ---------------------------------------------------------------------------
extern "C" void kernel_launch(void* const* d_in, const int* in_sizes, int n_in,
                              void* d_out, int out_size, void* d_ws, size_t ws_size,
                              hipStream_t stream) {
  const float* h_A = (const float*)d_in[0];
  const float* h_B = (const float*)d_in[1];
  const int* src_r[3] = {(const int*)d_in[2], (const int*)d_in[4], (const int*)d_in[6]};
  const int* dst_r[3] = {(const int*)d_in[3], (const int*)d_in[5], (const int*)d_in[7]};
  // A: 8..18  Wk bk Wq bq Wv bv Wa ba skip g be ; B: 19..29
  const float* Wk_A = (const float*)d_in[8];
  const float* bk_A = (const float*)d_in[9];
  const float* Wq_A = (const float*)d_in[10];
  const float* bq_A = (const float*)d_in[11];
  const float* Wv_A = (const float*)d_in[12];
  const float* bv_A = (const float*)d_in[13];
  const float* Wa_A = (const float*)d_in[14];
  const float* ba_A = (const float*)d_in[15];
  const float* skip_A = (const float*)d_in[16];
  const float* g_A = (const float*)d_in[17];
  const float* be_A = (const float*)d_in[18];
  const float* Wk_B = (const float*)d_in[19];
  const float* bk_B = (const float*)d_in[20];
  const float* Wq_B = (const float*)d_in[21];
  const float* bq_B = (const float*)d_in[22];
  const float* Wv_B = (const float*)d_in[23];
  const float* bv_B = (const float*)d_in[24];
  const float* Wa_B = (const float*)d_in[25];
  const float* ba_B = (const float*)d_in[26];
  const float* skip_B = (const float*)d_in[27];
  const float* g_B = (const float*)d_in[28];
  const float* be_B = (const float*)d_in[29];
  const float* pri_r[3] = {(const float*)d_in[30], (const float*)d_in[33], (const float*)d_in[36]};
  const float* att_r[3] = {(const float*)d_in[31], (const float*)d_in[34], (const float*)d_in[37]};
  const float* msg_r[3] = {(const float*)d_in[32], (const float*)d_in[35], (const float*)d_in[38]};

  const int Nn = in_sizes[0] / DMODEL;   // 32768
  const int nE = in_sizes[2];            // 524288
  const size_t ND = (size_t)Nn * DMODEL;

  // workspace carve
  float* ws = (float*)d_ws;
  size_t off = 0;
  auto take = [&](size_t n) { float* p = ws + off; off += n; return p; };
  float *wcK[3], *bcK[3], *wcV[3], *bcV[3];
  for (int r = 0; r < 3; ++r) {
    wcK[r] = take(65536); bcK[r] = take(256);
    wcV[r] = take(65536); bcV[r] = take(256);
  }
  float* QA = take(ND);
  float* QB = take(ND);
  float* PA = take(ND);
  float* PB = take(ND);
  float* kt = take(ND);
  float* vt = take(ND);
  float* aggA = take(ND);   // aggA and aggB contiguous
  float* aggB = take(ND);
  float* score = take((size_t)nE * 4);
  float* smax = take((size_t)Nn * 4);
  float* den = take((size_t)Nn * 4);

  // --- weight prep (fold att/msg into Wk/Wv, output-major) ---
  prep_combined<<<DMODEL, 256, 0, stream>>>(Wk_A, bk_A, att_r[0], wcK[0], bcK[0]);
  prep_combined<<<DMODEL, 256, 0, stream>>>(Wv_A, bv_A, msg_r[0], wcV[0], bcV[0]);
  prep_combined<<<DMODEL, 256, 0, stream>>>(Wk_B, bk_B, att_r[1], wcK[1], bcK[1]);
  prep_combined<<<DMODEL, 256, 0, stream>>>(Wv_B, bv_B, msg_r[1], wcV[1], bcV[1]);
  prep_combined<<<DMODEL, 256, 0, stream>>>(Wk_A, bk_A, att_r[2], wcK[2], bcK[2]);
  prep_combined<<<DMODEL, 256, 0, stream>>>(Wv_A, bv_A, msg_r[2], wcV[2], bcV[2]);

  // --- node projections (WMMA GEMMs); W row-major is already output-major ---
  dim3 ggrid(Nn / 64);
  gemm_wmma_f32<<<ggrid, 256, 0, stream>>>(h_A, Wq_A, bq_A, QA, Nn);
  gemm_wmma_f32<<<ggrid, 256, 0, stream>>>(h_B, Wq_B, bq_B, QB, Nn);
  gemm_wmma_f32<<<ggrid, 256, 0, stream>>>(h_A, Wa_A, ba_A, PA, Nn);
  gemm_wmma_f32<<<ggrid, 256, 0, stream>>>(h_B, Wa_B, ba_B, PB, Nn);

  // zero both aggregators (contiguous)
  {
    int n = (int)(2 * ND);
    fill_f32<<<(n + 255) / 256, 256, 0, stream>>>(aggA, 0.f, n);
  }

  // --- per-relation edge phase (sequential: K/V stay L2-resident) ---
  const float* srcFeat[3] = {h_A, h_B, h_A};
  const float* Qdst[3] = {QB, QA, QA};       // r1 -> B, r2 -> A, r3 -> A
  float* aggDst[3] = {aggB, aggA, aggA};
  for (int r = 0; r < 3; ++r) {
    int nh = Nn * 4;
    fill_f32<<<(nh + 255) / 256, 256, 0, stream>>>(smax, -3.0e38f, nh);
    fill_f32<<<(nh + 255) / 256, 256, 0, stream>>>(den, 0.f, nh);
    gemm_wmma_f32<<<ggrid, 256, 0, stream>>>(srcFeat[r], wcK[r], bcK[r], kt, Nn);
    gemm_wmma_f32<<<ggrid, 256, 0, stream>>>(srcFeat[r], wcV[r], bcV[r], vt, Nn);
    edge_score<<<nE / 8, 256, 0, stream>>>(Qdst[r], kt, src_r[r], dst_r[r],
                                           pri_r[r], score, smax, nE);
    edge_exp<<<(nE * 4) / 256, 256, 0, stream>>>(dst_r[r], score, smax, den, nE);
    edge_agg<<<nE / 8, 256, 0, stream>>>(vt, src_r[r], dst_r[r], score, den,
                                         aggDst[r], nE);
  }

  // --- finalize: gated residual + LayerNorm ---
  float* outA = (float*)d_out;
  float* outB = outA + ND;
  finalize_ln<<<Nn, 256, 0, stream>>>(aggA, 0.5f, PA, skip_A, g_A, be_A, outA);
  finalize_ln<<<Nn, 256, 0, stream>>>(aggB, 1.0f, PB, skip_B, g_B, be_B, outB);
}